// OutputMPNN_32315333935301
// MI455X (gfx1250) — compile-verified
//
#include <hip/hip_runtime.h>
#include <math.h>

typedef __attribute__((ext_vector_type(2))) float v2f;
typedef __attribute__((ext_vector_type(8))) float v8f;

#define B_      8
#define N_      256
#define CIN     128
#define NB      32
#define LAYERW  256
#define PI_F    3.14159265358979323846f

// single v_rcp_f32 (~1 ulp) instead of the full div-fixup sequence
static __device__ __forceinline__ float fast_rcp(float x) {
    return __builtin_amdgcn_rcpf(x);
}

// ---------------------------------------------------------------------------
// Fused message-passing kernel:
//   x_mp[b,a,c] = sum_x (phi(r_bax) . rad_w[:,c] + rad_b[c])
//                 * sigmoid((cut_rad[c]-r)/cut_width[c]) * edge_mask * x[b,x,c]
// One block = (b, 16-wide a tile). 8 waves, each wave owns 16 channels.
// GEMM (pairs x 32) @ (32 x 128) runs on V_WMMA_F32_16X16X4_F32.
// launch_bounds min-waves=4 caps VGPRs at 256: no s_set_vgpr_msb, 2 blocks/WGP.
// ---------------------------------------------------------------------------
__global__ __launch_bounds__(256, 4) void mp_kernel(
    const float* __restrict__ xin,               // (B,N,CIN)
    const float* __restrict__ norms,             // (B,N,N)
    const unsigned char* __restrict__ edge_mask, // (B,N,N) bool
    const float* __restrict__ radw,              // (NB,CIN)
    const float* __restrict__ radb,              // (CIN)
    const float* __restrict__ cutr,              // (CIN)
    const float* __restrict__ cutw,              // (CIN)
    float* __restrict__ xmp)                     // (B,N,CIN)
{
    __shared__ float phi[256 * 34];   // [a16][x16][32k], k-stride padded to 34
    __shared__ float featl[16 * 132]; // [x16][128c], padded
    __shared__ float rtile[16 * 17];  // [a16][x16]
    __shared__ float emtile[16 * 17]; // [a16][x16]

    const int t    = threadIdx.x;
    const int wv   = t >> 5;          // wave id (wave32)
    const int lane = t & 31;
    const int half = lane >> 4;
    const int ln   = lane & 15;
    const int c    = wv * 16 + ln;    // this lane's channel

    const int b  = blockIdx.x >> 4;
    const int a0 = (blockIdx.x & 15) << 4;

    // B-fragments of rad_w: K=32 as 8 chunks of 4.
    // 32-bit B 4x16 layout mirrors A: vgpr0 lanes0-15 K=0 / lanes16-31 K=2; vgpr1 K=1/K=3.
    v2f bfr[8];
#pragma unroll
    for (int kc = 0; kc < 8; ++kc) {
        bfr[kc].x = radw[(4 * kc + 0 + 2 * half) * CIN + c];
        bfr[kc].y = radw[(4 * kc + 1 + 2 * half) * CIN + c];
    }
    const float rb = radb[c];
    const float cr = cutr[c];
    const float iw = fast_rcp(cutw[c]);

    float acc[16];
#pragma unroll
    for (int i = 0; i < 16; ++i) acc[i] = 0.f;

    const int al_s = t >> 4;   // staging: pair (a,x) owned by this thread
    const int xl_s = t & 15;

    for (int xt = 0; xt < 16; ++xt) {
        const int x0 = xt << 4;

        // ---- stage r, edge mask, masked radial basis phi ----
        {
            const int gi   = (b * N_ + (a0 + al_s)) * N_ + (x0 + xl_s);
            const float r  = norms[gi];
            const float em = edge_mask[gi] ? 1.f : 0.f;
            rtile[al_s * 17 + xl_s]  = r;
            emtile[al_s * 17 + xl_s] = em;

            const float mb  = (em != 0.f && r > 0.f) ? 1.f : 0.f;
            const float rs  = (mb != 0.f) ? r : 1.f;
            const float inv = fast_rcp(rs);
            float pw[4];
            pw[0] = mb; pw[1] = mb * inv; pw[2] = pw[1] * inv; pw[3] = pw[2] * inv;
            float tg[8];
#pragma unroll
            for (int k = 0; k < 4; ++k) {
                float s, co;
                __sincosf(rs * (PI_F * (float)k), &s, &co);
                tg[k] = s; tg[4 + k] = co;
            }
            float* pp = &phi[(al_s * 16 + xl_s) * 34];
#pragma unroll
            for (int p = 0; p < 4; ++p)
#pragma unroll
                for (int q = 0; q < 8; ++q)
                    pp[p * 8 + q] = pw[p] * tg[q];   // basis index = p*8 + q
        }
        // ---- stage feature tile (16 x rows, 128 channels) ----
        {
            const int xr = t >> 4;
            const int cb = (t & 15) * 8;
            const float* src = &xin[(b * N_ + (x0 + xr)) * CIN + cb];
#pragma unroll
            for (int j = 0; j < 8; ++j) featl[xr * 132 + cb + j] = src[j];
        }
        __syncthreads();

        // ---- per wave: 16 'a' rows; each row = 16x16x4 WMMA x 8 over K=32 ----
#pragma unroll
        for (int al = 0; al < 16; ++al) {
            v8f d = {0.f, 0.f, 0.f, 0.f, 0.f, 0.f, 0.f, 0.f};
            const float* pp = &phi[(al * 16 + ln) * 34];
#pragma unroll
            for (int kc = 0; kc < 8; ++kc) {
                v2f a;
                a.x = pp[4 * kc + 0 + 2 * half];
                a.y = pp[4 * kc + 1 + 2 * half];
                d = __builtin_amdgcn_wmma_f32_16x16x4_f32(
                        false, a, false, bfr[kc], (short)0, d, false, false);
            }
            // D layout: vgpr j: lanes0-15 -> x=j, lanes16-31 -> x=j+8; lane -> channel
            float part = 0.f;
#pragma unroll
            for (int j = 0; j < 8; ++j) {
                const int xl    = j + 8 * half;
                const float r   = rtile[al * 17 + xl];
                const float em  = emtile[al * 17 + xl];
                const float sft = fast_rcp(1.f + __expf(-(cr - r) * iw));
                const float e   = (d[j] + rb) * sft * em;
                part += e * featl[xl * 132 + c];
            }
            acc[al] += part;
        }
        __syncthreads();
    }

    // halves hold partials over x=0..7 / x=8..15 for the same channel: combine
#pragma unroll
    for (int al = 0; al < 16; ++al) {
        float tot = acc[al] + __shfl_xor(acc[al], 16, 32);
        if (half == 0) xmp[(b * N_ + (a0 + al)) * CIN + c] = tot;
    }
}

// ---------------------------------------------------------------------------
// MLP kernel: y = leakyrelu([x_mp|x] @ w1 + b1) @ w2 + b2, masked by atom_mask
// 16 rows per block; h-GEMM (K=256) on WMMA; 2nd GEMM on WMMA for COUT=128,
// scalar for COUT=2.
// ---------------------------------------------------------------------------
template <int COUT>
__global__ __launch_bounds__(256) void mlp_kernel(
    const float* __restrict__ xmp,           // (B*N, CIN)
    const float* __restrict__ xin,           // (B*N, CIN)
    const float* __restrict__ w1,            // (2*CIN, LAYERW)
    const float* __restrict__ b1,            // (LAYERW)
    const float* __restrict__ w2,            // (LAYERW, COUT)
    const float* __restrict__ b2,            // (COUT)
    const unsigned char* __restrict__ amask, // (B*N) bool
    float* __restrict__ out)                 // (B*N, COUT)
{
    __shared__ float inl[16 * 260];
    __shared__ float hl[16 * 260];

    const int t = threadIdx.x;
    const int wv = t >> 5, lane = t & 31, half = lane >> 4, ln = lane & 15;
    const int row0 = blockIdx.x * 16;

    // stage concat([x_mp, x]) : 16 rows x 256
    {
        const int rr = t >> 4;
        const int cb = (t & 15) * 16;
#pragma unroll
        for (int j = 0; j < 16; ++j) {
            const int cc = cb + j;
            const float v = (cc < CIN) ? xmp[(row0 + rr) * CIN + cc]
                                       : xin[(row0 + rr) * CIN + (cc - CIN)];
            inl[rr * 260 + cc] = v;
        }
    }
    __syncthreads();

    // h = leakyrelu(in @ w1 + b1); each wave -> 2 column tiles of 16
#pragma unroll
    for (int ct = 0; ct < 2; ++ct) {
        const int n0 = wv * 32 + ct * 16;
        v8f d = {0.f, 0.f, 0.f, 0.f, 0.f, 0.f, 0.f, 0.f};
        for (int kc = 0; kc < 64; ++kc) {
            v2f a, bb;
            a.x  = inl[ln * 260 + 4 * kc + 0 + 2 * half];
            a.y  = inl[ln * 260 + 4 * kc + 1 + 2 * half];
            bb.x = w1[(4 * kc + 0 + 2 * half) * LAYERW + n0 + ln];
            bb.y = w1[(4 * kc + 1 + 2 * half) * LAYERW + n0 + ln];
            d = __builtin_amdgcn_wmma_f32_16x16x4_f32(
                    false, a, false, bb, (short)0, d, false, false);
        }
        const float bias = b1[n0 + ln];
#pragma unroll
        for (int j = 0; j < 8; ++j) {
            const int rr = j + 8 * half;
            float h = d[j] + bias;
            h = (h >= 0.f) ? h : 0.01f * h;
            hl[rr * 260 + n0 + ln] = h;
        }
    }
    __syncthreads();

    if (COUT >= 16) {
        // y = h @ w2 + b2; wave wv -> output columns [wv*16, wv*16+16)
        const int n0 = wv * 16;
        v8f d = {0.f, 0.f, 0.f, 0.f, 0.f, 0.f, 0.f, 0.f};
        for (int kc = 0; kc < 64; ++kc) {
            v2f a, bb;
            a.x  = hl[ln * 260 + 4 * kc + 0 + 2 * half];
            a.y  = hl[ln * 260 + 4 * kc + 1 + 2 * half];
            bb.x = w2[(4 * kc + 0 + 2 * half) * COUT + n0 + ln];
            bb.y = w2[(4 * kc + 1 + 2 * half) * COUT + n0 + ln];
            d = __builtin_amdgcn_wmma_f32_16x16x4_f32(
                    false, a, false, bb, (short)0, d, false, false);
        }
        const float bias = b2[n0 + ln];
#pragma unroll
        for (int j = 0; j < 8; ++j) {
            const int rr  = j + 8 * half;
            const int row = row0 + rr;
            const float y = amask[row] ? (d[j] + bias) : 0.f;
            out[row * COUT + n0 + ln] = y;
        }
    } else {
        // tiny output head (COUT == 2): 32 outputs, one thread each (wave 0)
        if (t < 16 * COUT) {
            const int rr = t / COUT, co = t % COUT;
            float s = b2[co];
            for (int k = 0; k < LAYERW; ++k)
                s += hl[rr * 260 + k] * w2[k * COUT + co];
            const int row = row0 + rr;
            out[row * COUT + co] = amask[row] ? s : 0.f;
        }
    }
}

// ---------------------------------------------------------------------------
extern "C" void kernel_launch(void* const* d_in, const int* in_sizes, int n_in,
                              void* d_out, int out_size, void* d_ws, size_t ws_size,
                              hipStream_t stream) {
    const float* features        = (const float*)d_in[0];
    const float* norms           = (const float*)d_in[1];
    const unsigned char* amask   = (const unsigned char*)d_in[2];
    const unsigned char* emask   = (const unsigned char*)d_in[3];

    const float* radw0 = (const float*)d_in[4];
    const float* radb0 = (const float*)d_in[5];
    const float* cutr0 = (const float*)d_in[6];
    const float* cutw0 = (const float*)d_in[7];
    const float* w1_0  = (const float*)d_in[8];
    const float* b1_0  = (const float*)d_in[9];
    const float* w2_0  = (const float*)d_in[10];
    const float* b2_0  = (const float*)d_in[11];

    const float* radw1 = (const float*)d_in[12];
    const float* radb1 = (const float*)d_in[13];
    const float* cutr1 = (const float*)d_in[14];
    const float* cutw1 = (const float*)d_in[15];
    const float* w1_1  = (const float*)d_in[16];
    const float* b1_1  = (const float*)d_in[17];
    const float* w2_1  = (const float*)d_in[18];
    const float* b2_1  = (const float*)d_in[19];

    float* xmp0 = (float*)d_ws;                    // (B,N,CIN)
    float* y0   = xmp0 + B_ * N_ * CIN;            // (B,N,CIN)
    float* xmp1 = y0 + B_ * N_ * CIN;              // (B,N,CIN)

    dim3 grid(B_ * (N_ / 16));   // 128
    dim3 blk(256);

    mp_kernel<<<grid, blk, 0, stream>>>(features, norms, emask,
                                        radw0, radb0, cutr0, cutw0, xmp0);
    mlp_kernel<CIN><<<grid, blk, 0, stream>>>(xmp0, features,
                                              w1_0, b1_0, w2_0, b2_0, amask, y0);
    mp_kernel<<<grid, blk, 0, stream>>>(y0, norms, emask,
                                        radw1, radb1, cutr1, cutw1, xmp1);
    mlp_kernel<2><<<grid, blk, 0, stream>>>(xmp1, y0,
                                            w1_1, b1_1, w2_1, b2_1, amask,
                                            (float*)d_out);
}